// WordRNN_32203664785710
// MI455X (gfx1250) — compile-verified
//
#include <hip/hip_runtime.h>
#include <cstdint>

#define VOCAB 32000
#define EMB   128
#define HID   128
#define BATCH 8
#define SEQ   1024
#define ROWS  (BATCH * SEQ)   // 8192

// +2 pad (even, keeps 8B alignment for float2 LDS loads) to break the
// 16-way bank conflict of the fp32 WMMA operand access pattern.
#define LPAD 130

typedef float v2f __attribute__((ext_vector_type(2)));
typedef float v8f __attribute__((ext_vector_type(8)));

// ---------------------------------------------------------------------------
// WMMA inner product over K=128 with A(16x128) and B(128x128) staged in LDS.
// A layout per ISA: lane m=lane%16, group g=lane/16 holds K={2g,2g+1} per step.
// B layout mirrors it with n=lane%16.  D = A x B accumulated over 32 k-steps
// of V_WMMA_F32_16X16X4_F32.
// ---------------------------------------------------------------------------
__device__ __forceinline__ v8f wmma_k128(const float* __restrict__ Alds,
                                         const float* __restrict__ Brow,
                                         int m, int g) {
    v8f c = {};
#pragma unroll
    for (int kk = 0; kk < 32; ++kk) {
        const int k0 = kk * 4 + g * 2;
        v2f a = *(const v2f*)(Alds + m * LPAD + k0);
        v2f b = *(const v2f*)(Brow + k0);
        c = __builtin_amdgcn_wmma_f32_16x16x4_f32(
            /*neg_a=*/false, a, /*neg_b=*/false, b,
            /*c_mod=*/(short)0, c, /*reuse_a=*/false, /*reuse_b=*/false);
    }
    return c;
}

// ---------------------------------------------------------------------------
// Stage 1: x_proj[r,h] = dot(emb[x[r],:], W_ih[h,:]) + b_ih[h] + b_hh[h]
// Block: 256 threads = 8 waves; 16 rows x 128 cols per block.
// ---------------------------------------------------------------------------
__global__ __launch_bounds__(256) void embed_proj_kernel(
    const int* __restrict__ x, const float* __restrict__ emb,
    const float* __restrict__ W_ih, const float* __restrict__ b_ih,
    const float* __restrict__ b_hh, float* __restrict__ xp)
{
    __shared__ float A[16 * LPAD];
    __shared__ float Bsh[128 * LPAD];
    const int t    = threadIdx.x;
    const int row0 = blockIdx.x * 16;

    // Gather 16 embedding rows into LDS (coalesced within each row).
#pragma unroll
    for (int i = 0; i < 8; ++i) {
        const int idx = t + i * 256;
        const int m = idx >> 7, k = idx & 127;
        const int tok = x[row0 + m];
        A[m * LPAD + k] = emb[(size_t)tok * EMB + k];
    }
    // Load all of W_ih (64 KB) into LDS, coalesced.
#pragma unroll 8
    for (int i = 0; i < 64; ++i) {
        const int idx = t + i * 256;
        Bsh[(idx >> 7) * LPAD + (idx & 127)] = W_ih[idx];
    }
    __syncthreads();

    const int wave = t >> 5, lane = t & 31;
    const int n = lane & 15, g = lane >> 4;
    const int col = wave * 16 + n;            // hidden unit 0..127

    v8f c = wmma_k128(A, Bsh + col * LPAD, n, g);

    const float bias = b_ih[col] + b_hh[col];
#pragma unroll
    for (int r = 0; r < 8; ++r) {
        const int m = r + 8 * g;
        xp[(size_t)(row0 + m) * HID + col] = c[r] + bias;
    }
}

// ---------------------------------------------------------------------------
// Stage 2: sequential RNN scan.  One 1024-thread block (32 waves).
// thread t -> (b = t/128, j = t%128).  W_hh LDS-resident (stride 129 =>
// conflict-free column reads), h double-buffered in LDS, 1 barrier/step.
// ---------------------------------------------------------------------------
__global__ __launch_bounds__(1024) void rnn_scan_kernel(
    const float* __restrict__ xp, const float* __restrict__ W_hh,
    float* __restrict__ hs, float* __restrict__ h_last)
{
    __shared__ float w[HID * (HID + 1)];        // 128x129 floats
    __shared__ float hbuf[2][BATCH * HID];      // double-buffered h
    const int t = threadIdx.x;
    const int b = t >> 7;
    const int j = t & 127;

    for (int i = t; i < HID * HID; i += 1024)
        w[(i >> 7) * (HID + 1) + (i & 127)] = W_hh[i];
    hbuf[0][t] = 0.0f;
    __syncthreads();

    int cur = 0;
    for (int s = 0; s < SEQ; ++s) {
        float acc = xp[((size_t)b * SEQ + s) * HID + j];
        const float* hr = &hbuf[cur][b * HID];  // broadcast reads
        const float* wr = &w[j * (HID + 1)];    // conflict-free
#pragma unroll 16
        for (int k = 0; k < HID; ++k) acc += hr[k] * wr[k];
        const float hn = tanhf(acc);
        hs[((size_t)b * SEQ + s) * HID + j] = hn;
        hbuf[cur ^ 1][t] = hn;
        __syncthreads();
        cur ^= 1;
    }
    h_last[t] = hbuf[cur][t];                   // [1,B,H] tail of d_out
}

// ---------------------------------------------------------------------------
// Stage 3 (dominant): logits[r,v] = dot(hs[r,:], fc_W[v,:]) + fc_b[v]
// Grid: (V/128, ROWS/16).  Block: 256 threads = 8 waves -> 16x128 tile.
// fc_W slab for the block (128x128) staged in LDS; 256 WMMA f32 per block.
// ---------------------------------------------------------------------------
__global__ __launch_bounds__(256) void logits_kernel(
    const float* __restrict__ hs, const float* __restrict__ fcW,
    const float* __restrict__ fcb, float* __restrict__ out)
{
    __shared__ float A[16 * LPAD];
    __shared__ float Bsh[128 * LPAD];
    const int t       = threadIdx.x;
    const int row0    = blockIdx.y * 16;
    const int colbase = blockIdx.x * 128;

#pragma unroll
    for (int i = 0; i < 8; ++i) {
        const int idx = t + i * 256;
        const int m = idx >> 7, k = idx & 127;
        A[m * LPAD + k] = hs[(size_t)(row0 + m) * HID + k];
    }
    const float* Wblk = fcW + (size_t)colbase * HID;   // 128 contiguous rows
#pragma unroll 8
    for (int i = 0; i < 64; ++i) {
        const int idx = t + i * 256;
        Bsh[(idx >> 7) * LPAD + (idx & 127)] = Wblk[idx];
    }
    __syncthreads();

    const int wave = t >> 5, lane = t & 31;
    const int n = lane & 15, g = lane >> 4;
    const int col = colbase + wave * 16 + n;

    v8f c = wmma_k128(A, Bsh + (wave * 16 + n) * LPAD, n, g);

    const float bias = fcb[col];
#pragma unroll
    for (int r = 0; r < 8; ++r) {
        const int m = r + 8 * g;
        out[(size_t)(row0 + m) * VOCAB + col] = c[r] + bias;
    }
}

// ---------------------------------------------------------------------------
extern "C" void kernel_launch(void* const* d_in, const int* in_sizes, int n_in,
                              void* d_out, int out_size, void* d_ws, size_t ws_size,
                              hipStream_t stream) {
    (void)in_sizes; (void)n_in; (void)out_size; (void)ws_size;
    const int*   x     = (const int*)d_in[0];
    const float* emb   = (const float*)d_in[1];
    const float* W_ih  = (const float*)d_in[2];
    const float* b_ih  = (const float*)d_in[3];
    const float* W_hh  = (const float*)d_in[4];
    const float* b_hh  = (const float*)d_in[5];
    const float* fcW   = (const float*)d_in[6];
    const float* fcb   = (const float*)d_in[7];

    float* logits = (float*)d_out;
    float* h_last = logits + (size_t)ROWS * VOCAB;   // [1,B,H] appended

    float* xp = (float*)d_ws;                        // [ROWS, HID] = 4 MB
    float* hs = xp + (size_t)ROWS * HID;             // [ROWS, HID] = 4 MB

    embed_proj_kernel<<<dim3(ROWS / 16), 256, 0, stream>>>(x, emb, W_ih, b_ih, b_hh, xp);
    rnn_scan_kernel<<<dim3(1), 1024, 0, stream>>>(xp, W_hh, hs, h_last);
    logits_kernel<<<dim3(VOCAB / 128, ROWS / 16), 256, 0, stream>>>(hs, fcW, fcb, logits);
}